// BPMLLLoss_64596308132130
// MI455X (gfx1250) — compile-verified
//
#include <hip/hip_runtime.h>
#include <math.h>

// BP-MLL loss, MI455X (gfx1250, wave32).
// Stage 1: one block per row, streaming NT b128 loads, per-row value -> d_ws.
// Stage 2: single block reduces 8192 row values; wave-level sum done via
//          v_wmma_f32_16x16x4_f32 with B = ones (exact fp32 cross-lane add).

typedef __attribute__((ext_vector_type(4))) float v4f;
typedef __attribute__((ext_vector_type(4))) int   v4i;
typedef __attribute__((ext_vector_type(2))) float v2f;
typedef __attribute__((ext_vector_type(8))) float v8f;

#define LROW 10000
#define NROW 8192
#define TPB  256

__global__ __launch_bounds__(TPB)
void bpmll_rows(const float* __restrict__ x, const int* __restrict__ tgt,
                float* __restrict__ rowval) {
  const int row = blockIdx.x;
  const int tid = threadIdx.x;
  const v4f* __restrict__ xr = (const v4f*)(x   + (size_t)row * LROW);
  const v4i* __restrict__ tr = (const v4i*)(tgt + (size_t)row * LROW);

  float sp = 0.f, sn = 0.f, np = 0.f;
  // LROW/4 = 2500 b128 chunks per row, strided across 256 threads.
  for (int i = tid; i < LROW / 4; i += TPB) {
    v4f xv = __builtin_nontemporal_load(xr + i);   // global_load_b128 nt
    v4i tv = __builtin_nontemporal_load(tr + i);   // global_load_b128 nt
#pragma unroll
    for (int j = 0; j < 4; ++j) {
      const bool  p = (tv[j] != 0);
      const float e = expf(p ? -xv[j] : xv[j]);    // one v_exp_f32 path per elem
      sp += p ? e   : 0.0f;
      sn += p ? 0.f : e;
      np += p ? 1.f : 0.0f;
    }
  }

  // wave32 tree reduction
#pragma unroll
  for (int m = 16; m >= 1; m >>= 1) {
    sp += __shfl_xor(sp, m, 32);
    sn += __shfl_xor(sn, m, 32);
    np += __shfl_xor(np, m, 32);
  }

  __shared__ float rsp[8], rsn[8], rnp[8];
  const int wave = tid >> 5, lane = tid & 31;
  if (lane == 0) { rsp[wave] = sp; rsn[wave] = sn; rnp[wave] = np; }
  __syncthreads();

  if (tid < 32) {
    float a = (lane < 8) ? rsp[lane] : 0.f;
    float b = (lane < 8) ? rsn[lane] : 0.f;
    float c = (lane < 8) ? rnp[lane] : 0.f;
#pragma unroll
    for (int m = 4; m >= 1; m >>= 1) {
      a += __shfl_xor(a, m, 32);
      b += __shfl_xor(b, m, 32);
      c += __shfl_xor(c, m, 32);
    }
    if (lane == 0) {
      const float k = c * ((float)LROW - c);       // |Y| * |Ybar|
      rowval[row] = a * b / k;
    }
  }
}

__global__ __launch_bounds__(256)
void bpmll_final(const float* __restrict__ rowval, float* __restrict__ out) {
  const int tid = threadIdx.x;

  float s = 0.f;
  for (int i = tid; i < NROW; i += 256) s += rowval[i];

  __shared__ float part[256];
  part[tid] = s;
  __syncthreads();

  // Every wave redundantly computes the same total (EXEC all-ones at the WMMA).
  const int lane = tid & 31;
  float v = 0.f;
#pragma unroll
  for (int j = 0; j < 8; ++j) v += part[lane * 8 + j];

  // WMMA cross-lane sum: A[m][k] = [v[m], 0, v[m+16], 0], B = ones(4x16)
  // => D[m][n] = v[m] + v[m+16] (layout of ones-B is irrelevant).
  v2f a; a[0] = v;   a[1] = 0.0f;
  v2f b; b[0] = 1.f; b[1] = 1.0f;
  v8f c = {};
  c = __builtin_amdgcn_wmma_f32_16x16x4_f32(
      /*neg_a=*/false, a, /*neg_b=*/false, b,
      /*c_mod=*/(short)0, c, /*reuse_a=*/false, /*reuse_b=*/false);

  float t = 0.f;
#pragma unroll
  for (int r = 0; r < 8; ++r) t += c[r];
  // lanes 0..15 hold sum(v0..7 + v16..23); lanes 16..31 hold sum(v8..15 + v24..31)
  const float total = __shfl(t, 0, 32) + __shfl(t, 16, 32);

  if (tid == 0) out[0] = total;
}

extern "C" void kernel_launch(void* const* d_in, const int* in_sizes, int n_in,
                              void* d_out, int out_size, void* d_ws, size_t ws_size,
                              hipStream_t stream) {
  const float* x   = (const float*)d_in[0];
  const int*   tgt = (const int*)d_in[1];
  float* rowval    = (float*)d_ws;   // NROW floats = 32 KB scratch

  bpmll_rows<<<NROW, TPB, 0, stream>>>(x, tgt, rowval);
  bpmll_final<<<1, 256, 0, stream>>>(rowval, (float*)d_out);
}